// EfficientIoU_10067403341971
// MI455X (gfx1250) — compile-verified
//
#include <hip/hip_runtime.h>
#include <math.h>

// ---------------------------------------------------------------------------
// EIoU + BCE fused loss, mean over N boxes (memory-bound: 168 MB @ 23.3 TB/s
// => ~7 us floor; ~0.25 GFLOP total is noise).
//   pass 1: double-buffered async streaming (global_load_async_to_lds_b128,
//           partial s_wait_asynccnt) of 512-box tiles of preds/target into
//           LDS; per-box loss on VALU; deterministic block reduction.
//   pass 2: single wave sums block partials; the final 32-value collapse runs
//           on the matrix pipe via V_WMMA_F32_16X16X4_F32 with a ones operand.
// ---------------------------------------------------------------------------

#define BLOCK        256
#define GRID         1024
#define TILE_BOXES   512
#define TILE_FLOATS  (TILE_BOXES * 5)          // 2560 floats per array tile
#define TILE_BYTES   (TILE_FLOATS * 4)         // 10240 bytes per array tile
#define XFERS_PER_AR (TILE_BYTES / 16)         // 640 b128 transfers per array
#define XFERS_TOTAL  (2 * XFERS_PER_AR)        // 1280 per tile => 5 / thread
#define BUF_FLOATS   (2 * TILE_FLOATS)         // preds + target regions
#define BCE_COEFF    0.2f

typedef __attribute__((ext_vector_type(2))) float v2f;
typedef __attribute__((ext_vector_type(8))) float v8f;

__device__ __forceinline__ float sigmoid_f32(float z) {
  return 1.0f / (1.0f + expf(-z));
}

// Full per-box loss contribution: eiou_i + BCE_COEFF * bce_i
__device__ __forceinline__ float box_loss(float plogit, float pa, float pb,
                                          float pc, float pd,
                                          float ty, float t0, float t1,
                                          float t2, float t3) {
  const float p0 = sigmoid_f32(pa);
  const float p1 = sigmoid_f32(pb);
  const float p2 = sigmoid_f32(pc);
  const float p3 = sigmoid_f32(pd);

  const float pred_area   = fabsf(p2 - p0) * fabsf(p3 - p1);
  const float target_area = (t2 - t0) * (t3 - t1);

  const float x_p1 = fminf(p2, p0), x_p2 = fmaxf(p2, p0);
  const float y_p1 = fminf(p1, p3), y_p2 = fmaxf(p1, p3);

  const float x1 = fmaxf(x_p1, t0), x2 = fminf(x_p2, t2);
  const float y1 = fmaxf(y_p1, t1), y2 = fminf(y_p2, t3);
  float overlap = (x2 - x1) * (y2 - y1);
  overlap = overlap < 0.0f ? 0.0f : overlap;

  const float x1c = fminf(x_p1, t0), x2c = fmaxf(x_p2, t2);
  const float y1c = fminf(y_p1, t1), y2c = fmaxf(y_p2, t3);
  const float c_w = x2c - x1c;
  const float c_h = y2c - y1c;

  const float w   = x_p2 - x_p1, w_t = t2 - t0;
  const float h   = y_p2 - y_p1, h_t = t3 - t1;

  const float iou = overlap / (target_area + pred_area - overlap);

  const float cpx = (x_p2 + x_p1) * 0.5f, cpy = (y_p2 + y_p1) * 0.5f;
  const float ctx = (t2 + t0) * 0.5f,     cty = (t3 + t1) * 0.5f;
  const float diag = c_w * c_w + c_h * c_h;
  const float center = ((cpx - ctx) * (cpx - ctx) +
                        (cpy - cty) * (cpy - cty)) / diag;
  const float width  = (w - w_t) * (w - w_t) / (c_w * c_w);
  const float height = (h - h_t) * (h - h_t) / (h_t * h_t);

  const float eiou = 1.0f - (iou - (center + width + height));

  const float x = plogit;
  const float bce = fmaxf(x, 0.0f) - x * ty + log1pf(expf(-fabsf(x)));

  return eiou + BCE_COEFF * bce;
}

__device__ __forceinline__ float wave_reduce_add(float v) {
#pragma unroll
  for (int off = 16; off > 0; off >>= 1)
    v += __shfl_xor(v, off, 32);
  return v;
}

__global__ void __launch_bounds__(BLOCK)
eiou_partials(const float* __restrict__ preds,
              const float* __restrict__ target,
              float* __restrict__ partials, int n) {
  // Two 20 KB buffers: [preds tile | target tile] each. 40 KB total; trivial
  // against the 320 KB WGP LDS pool.
  __shared__ __align__(16) float tiles[2][BUF_FLOATS];
  __shared__ float waveSums[BLOCK / 32];

  const int tid       = threadIdx.x;
  const int stride    = (int)gridDim.x;
  const int fullTiles = n / TILE_BOXES;
  float acc = 0.0f;

  // Issue one tile's async copies: exactly XFERS_TOTAL/BLOCK = 5 b128 async
  // instructions per thread (uniform per wave -> uniform ASYNCcnt).
  auto issue_tile = [&](int tile, int buf) {
    const unsigned ldsBase  = (unsigned)(size_t)(&tiles[buf][0]);
    const size_t   baseByte = (size_t)tile * (size_t)TILE_BYTES;
    const unsigned long long pBase =
        (unsigned long long)(size_t)preds + baseByte;
    const unsigned long long tBase =
        (unsigned long long)(size_t)target + baseByte;
#pragma unroll
    for (int x = tid; x < XFERS_TOTAL; x += BLOCK) {
      const unsigned lds = ldsBase + (unsigned)x * 16u;  // target region at
                                                         // x>=640 lands at
                                                         // ldsBase+10240: ok
      const unsigned long long g =
          (x < XFERS_PER_AR) ? (pBase + (size_t)x * 16u)
                             : (tBase + (size_t)(x - XFERS_PER_AR) * 16u);
      asm volatile("global_load_async_to_lds_b128 %0, %1, off"
                   :: "v"(lds), "v"(g)
                   : "memory");
    }
  };

  // Prime the pipeline.
  if ((int)blockIdx.x < fullTiles) issue_tile((int)blockIdx.x, 0);

  int buf = 0;
  for (int k = (int)blockIdx.x; k < fullTiles; k += stride) {
    const int nk = k + stride;
    if (nk < fullTiles) {
      issue_tile(nk, buf ^ 1);               // prefetch next tile
      // Wait only for the CURRENT tile's 5 per-wave transfers; leave the
      // prefetched tile's 5 in flight (copy/compute overlap).
      asm volatile("s_wait_asynccnt 0x5" ::: "memory");
    } else {
      asm volatile("s_wait_asynccnt 0x0" ::: "memory");
    }
    __syncthreads();  // all waves' current-tile data visible in LDS

    // 2 boxes per thread; 5*box addressing is bank-conflict-free (gcd(5,64)=1)
    const float* bp = &tiles[buf][0];
#pragma unroll
    for (int b = tid; b < TILE_BOXES; b += BLOCK) {
      const float* p = bp + b * 5;
      const float* t = bp + TILE_FLOATS + b * 5;
      acc += box_loss(p[0], p[1], p[2], p[3], p[4],
                      t[0], t[1], t[2], t[3], t[4]);
    }
    __syncthreads();  // this buffer may be rewritten by next-next prefetch
    buf ^= 1;
  }

  // Tail boxes (n % TILE_BOXES), block 0 with plain loads.
  if (blockIdx.x == 0) {
    for (int i = fullTiles * TILE_BOXES + tid; i < n; i += BLOCK) {
      const float* p = preds + (size_t)i * 5;
      const float* t = target + (size_t)i * 5;
      acc += box_loss(p[0], p[1], p[2], p[3], p[4],
                      t[0], t[1], t[2], t[3], t[4]);
    }
  }

  // Deterministic block reduction: wave32 shuffle tree -> LDS -> thread 0.
  const float wsum = wave_reduce_add(acc);
  if ((tid & 31) == 0) waveSums[tid >> 5] = wsum;
  __syncthreads();
  if (tid == 0) {
    float b = 0.0f;
#pragma unroll
    for (int w = 0; w < BLOCK / 32; ++w) b += waveSums[w];
    partials[blockIdx.x] = b;
  }
}

// Single-wave finisher. Each lane serially sums a strided slice of the block
// partials, then the 32 lane values (a 16x4 f32 A-matrix, 2 VGPRs) collapse
// on the matrix pipe: D = A x ones(4x16); column 0 of D is the row-sum
// vector, so lane0 + lane16's 8 accumulator VGPRs sum to the grand total.
__global__ void __launch_bounds__(32)
final_reduce(const float* __restrict__ partials, int nPartials,
             float* __restrict__ out, float invN) {
  float s = 0.0f;
  for (int i = (int)threadIdx.x; i < nPartials; i += 32) s += partials[i];

  v2f a;  a.x = s;    a.y = 0.0f;   // one live A value per lane
  v2f b;  b.x = 1.0f; b.y = 1.0f;   // B: 4x16 all-ones
  v8f c = {};
  c = __builtin_amdgcn_wmma_f32_16x16x4_f32(
        /*neg_a=*/false, a, /*neg_b=*/false, b,
        /*c_mod=*/(short)0, c, /*reuse_a=*/false, /*reuse_b=*/false);

  float colsum = c[0] + c[1] + c[2] + c[3] + c[4] + c[5] + c[6] + c[7];
  const float other = __shfl(colsum, 16, 32);
  if (threadIdx.x == 0) out[0] = (colsum + other) * invN;
}

extern "C" void kernel_launch(void* const* d_in, const int* in_sizes, int n_in,
                              void* d_out, int out_size, void* d_ws,
                              size_t ws_size, hipStream_t stream) {
  const float* preds  = (const float*)d_in[0];
  const float* target = (const float*)d_in[1];
  float* out      = (float*)d_out;
  float* partials = (float*)d_ws;   // GRID floats of scratch

  const int n = in_sizes[0] / 5;    // number of boxes

  eiou_partials<<<GRID, BLOCK, 0, stream>>>(preds, target, partials, n);
  final_reduce<<<1, 32, 0, stream>>>(partials, GRID, out, 1.0f / (float)n);
}